// CommitmentScoringAttn_57303453663918
// MI455X (gfx1250) — compile-verified
//
#include <hip/hip_runtime.h>
#include <hip/hip_bf16.h>

// Problem constants (from reference setup_inputs)
#define B_  8
#define H_  16
#define M_  1024
#define D_  64
#define LK_ 4096

#define SPLITS 16
#define ROWS_PER_SPLIT (LK_ / SPLITS)   // 256

typedef __attribute__((ext_vector_type(2))) float v2f;
typedef __attribute__((ext_vector_type(8))) float v8f;

// ---------------------------------------------------------------------------
// Phase 1: key_sum partials.  key [B,H,LK,D] -> part [B*H, SPLITS, D]
// 2048 blocks x 256 threads, coalesced float4 streaming (134 MB pass).
// ---------------------------------------------------------------------------
__global__ void __launch_bounds__(256)
ksum_partial_kernel(const float* __restrict__ key, float* __restrict__ part) {
    __shared__ float4 red[256];

    const int t  = threadIdx.x;
    const int bh = blockIdx.x / SPLITS;     // 0..127
    const int sp = blockIdx.x % SPLITS;     // 0..15
    const int c4 = t & 15;                  // float4 column within a 64-float row
    const int rg = t >> 4;                  // row group 0..15

    const float4* k4 = (const float4*)key;
    const long rowBase = (long)bh * LK_ + (long)sp * ROWS_PER_SPLIT;

    float4 acc; acc.x = 0.f; acc.y = 0.f; acc.z = 0.f; acc.w = 0.f;
    #pragma unroll 4
    for (int r = rg; r < ROWS_PER_SPLIT; r += 16) {
        float4 v = k4[(rowBase + r) * (D_ / 4) + c4];
        acc.x += v.x; acc.y += v.y; acc.z += v.z; acc.w += v.w;
    }
    red[t] = acc;
    __syncthreads();

    // tree-reduce over the 16 row groups (stride 16 in the LDS index)
    #pragma unroll
    for (int s = 8; s >= 1; s >>= 1) {
        if (rg < s) {
            float4 o = red[t + s * 16];
            red[t].x += o.x; red[t].y += o.y; red[t].z += o.z; red[t].w += o.w;
        }
        __syncthreads();
    }

    if (rg == 0) {
        ((float4*)part)[(long)blockIdx.x * (D_ / 4) + c4] = red[t];
    }
}

// ---------------------------------------------------------------------------
// Phase 2: logits = Q . key_sum via V_WMMA_F32_16X16X4_F32, then sigmoid.
// Grid: (B*H) * 8 m-tiles of 128 rows; block = 256 threads = 8 waves,
// each wave produces 16 rows of M with 16 chained f32 WMMAs (K=4 each).
// B operand = key_sum broadcast into all 16 columns -> every column of C
// holds the logit (redundant compute; we are far under the compute roof).
// ---------------------------------------------------------------------------
__global__ void __launch_bounds__(256)
qdot_sigmoid_kernel(const float* __restrict__ q,
                    const float* __restrict__ bias,
                    const float* __restrict__ part,
                    float* __restrict__ out) {
    __shared__ float ksum[D_];

    const int t    = threadIdx.x;
    const int bh   = blockIdx.x >> 3;   // 0..127
    const int mt   = blockIdx.x & 7;    // m-tile (128 rows each)
    const int lane = t & 31;
    const int w    = t >> 5;            // wave 0..7

    // reduce the SPLITS partials into key_sum (tiny: 16*64 floats)
    if (t < D_) {
        float s = 0.f;
        #pragma unroll
        for (int sp = 0; sp < SPLITS; ++sp)
            s += part[((long)bh * SPLITS + sp) * D_ + t];
        ksum[t] = s;
    }
    __syncthreads();

    const int m0   = mt * 128 + w * 16;        // first M row of this wave's tile
    const int mrow = m0 + (lane & 15);         // A-matrix row held by this lane
    const int koff = (lane >> 4) * 2;          // lanes 0-15: K=0,1  lanes 16-31: K=2,3

    const float* qrow = q + ((long)bh * M_ + mrow) * D_;

    v8f c = {};  // accumulator starts at 0
    #pragma unroll
    for (int s = 0; s < 16; ++s) {
        // A: 16x4 f32 tile of Q (ISA 7.12.2 layout), one b64 load per lane
        v2f a = *(const v2f*)(qrow + 4 * s + koff);
        // B: 4x16 with every column = key_sum chunk (broadcast ds_load_b64)
        v2f b = *(const v2f*)(ksum + 4 * s + koff);
        c = __builtin_amdgcn_wmma_f32_16x16x4_f32(
                /*neg_a=*/false, a, /*neg_b=*/false, b,
                /*c_mod=*/(short)0, c, /*reuse_a=*/false, /*reuse_b=*/false);
    }

    // C/D layout: VGPR r, lanes 0-15 -> M=r, lanes 16-31 -> M=8+r (all N equal)
    const int h = bh & (H_ - 1);
    float* o = out + (long)bh * M_;
    const float* bb = bias + (long)h * M_;

    if (lane == 0) {
        #pragma unroll
        for (int r = 0; r < 8; ++r) {
            int m = m0 + r;
            float x = c[r] + bb[m];
            o[m] = 1.0f / (1.0f + expf(-x));
        }
    } else if (lane == 16) {
        #pragma unroll
        for (int r = 0; r < 8; ++r) {
            int m = m0 + 8 + r;
            float x = c[r] + bb[m];
            o[m] = 1.0f / (1.0f + expf(-x));
        }
    }
}

// ---------------------------------------------------------------------------
extern "C" void kernel_launch(void* const* d_in, const int* in_sizes, int n_in,
                              void* d_out, int out_size, void* d_ws, size_t ws_size,
                              hipStream_t stream) {
    const float* query = (const float*)d_in[0];  // [B,H,M,D]
    const float* key   = (const float*)d_in[1];  // [B,H,LK,D]
    const float* bias  = (const float*)d_in[2];  // [H,M]
    float* out  = (float*)d_out;                 // [B,H,M,1] flat
    float* part = (float*)d_ws;                  // [B*H, SPLITS, D] = 512 KB

    ksum_partial_kernel<<<B_ * H_ * SPLITS, 256, 0, stream>>>(key, part);
    qdot_sigmoid_kernel<<<B_ * H_ * 8, 256, 0, stream>>>(query, bias, part, out);
}